// LSTM_12060268167556
// MI455X (gfx1250) — compile-verified
//
#include <hip/hip_runtime.h>
#include <math.h>

// LSTM scan, T=1024, B=64, H=512, fp32, gate order i,f,g,o, reset -> zero carry.
// Persistent kernel: 32 blocks x 128 thr = 128 wave32 => 1 wave per SIMD on 32
// WGPs (minimizes per-SIMD serialization of the WMMA stream). Device-scope
// atomic barrier between steps. Each wave computes one 16x16 batch/column tile
// in ALL FOUR gate blocks via V_WMMA_F32_16X16X4_F32 over fused K = 512(x@Wi)
// + 512(h@Wh), with split x/h accumulators (8 independent WMMA chains).
// Weights are pre-packed once per launch into per-lane fragment order so the
// hot loop feeds B with coalesced global_load_b128.

namespace {

constexpr int kT = 1024;
constexpr int kB = 64;
constexpr int kH = 512;
constexpr int kH4 = 4 * kH;
constexpr int kBlocks = 32;    // 4 row-tiles x 8 col-groups
constexpr int kThreads = 128;  // 4 wave32 per block -> 1 wave/SIMD
constexpr int kPackElems = kH * kH4;  // 1,048,576 floats per matrix

typedef float v2f __attribute__((ext_vector_type(2)));
typedef float v4f __attribute__((ext_vector_type(4)));
typedef float v8f __attribute__((ext_vector_type(8)));

__device__ __forceinline__ float sigmoid_f(float x) {
  return 1.0f / (1.0f + __expf(-x));
}

__device__ __forceinline__ float fast_tanh(float x) {
#if __has_builtin(__builtin_amdgcn_tanhf)
  return __builtin_amdgcn_tanhf(x);  // V_TANH_F32, single TRANS op
#else
  const float e = __expf(2.0f * x);  // overflow->inf gives correct +/-1 limits
  return 1.0f - 2.0f / (e + 1.0f);
#endif
}

// Pack W[H,4H] row-major into fragment order:
//   packed[(((nt*64 + k8)*4 + g)*32 + lane)*4 + v] = W[row][col]
//   row = k8*8 + 4*(v>>1) + 2*(lane>>4) + (v&1),  col = g*512 + nt*16 + (lane&15)
// so each lane's b128 load yields {B[k],B[k+1]} and {B[k+4],B[k+5]} frags
// for two consecutive WMMA k-steps of one gate.
__global__ void pack_weights_kernel(const float* __restrict__ Wi,
                                    const float* __restrict__ Wh,
                                    float* __restrict__ pWi,
                                    float* __restrict__ pWh) {
  const int i = blockIdx.x * blockDim.x + threadIdx.x;
  if (i >= kPackElems) return;
  const int v    = i & 3;
  const int lane = (i >> 2) & 31;
  const int g    = (i >> 7) & 3;
  const int k8   = (i >> 9) & 63;
  const int nt   = (i >> 15) & 31;
  const int row  = k8 * 8 + ((v >> 1) << 2) + 2 * (lane >> 4) + (v & 1);
  const int col  = g * kH + nt * 16 + (lane & 15);
  pWi[i] = Wi[(size_t)row * kH4 + col];
  pWh[i] = Wh[(size_t)row * kH4 + col];
}

__global__ __launch_bounds__(kThreads, 1)
void lstm_scan_kernel(const float* __restrict__ h0,
                      const float* __restrict__ c0,
                      const float* __restrict__ ins,     // [T,B,H]
                      const int*   __restrict__ resets,  // [T,B] (0/1)
                      const float* __restrict__ pWi,     // packed Wi
                      const float* __restrict__ pWh,     // packed Wh
                      const float* __restrict__ bias,    // [4H]
                      float*       __restrict__ ys,      // [T,B,H]
                      float*       __restrict__ c_ws,    // [B,H]
                      int*         __restrict__ arrive)  // grid barrier counter
{
  __shared__ float lds_x[16 * kH];  // 32 KB: x_t rows m0..m0+15
  __shared__ float lds_h[16 * kH];  // 32 KB: reset-masked h rows m0..m0+15
  __shared__ int   lds_r[16];       // reset flags for this block's batch rows

  const int tid  = threadIdx.x;
  const int bx   = blockIdx.x;
  const int wave = tid >> 5;
  const int lane = tid & 31;

  const int mt = bx & 3;                 // batch tile 0..3 (16 rows each)
  const int nt = (bx >> 2) * 4 + wave;   // column tile 0..31 within H
  const int m0 = mt * 16;
  const int j0 = nt * 16;

  const int am = lane & 15;  // A row (M); also D column index N
  const int kh = lane >> 4;  // K half-select for A/B; row-half for D

  // per-wave packed-weight bases (lane folded in); k8/gate offsets are static
  const float* pwi = pWi + ((size_t)nt << 15) + lane * 4;
  const float* pwh = pWh + ((size_t)nt << 15) + lane * 4;

  v4f* lds_x4 = (v4f*)lds_x;
  v4f* lds_h4 = (v4f*)lds_h;

  for (int t = 0; t < kT; ++t) {
    const float* x_t    = ins + (size_t)t * kB * kH;
    const int*   r_t    = resets + (size_t)t * kB;
    const float* h_prev = (t == 0) ? h0 : ys + (size_t)(t - 1) * kB * kH;
    const float* c_prev = (t == 0) ? c0 : c_ws;
    float*       h_out  = ys + (size_t)t * kB * kH;

    // ---- stage reset flags, x tile and reset-masked h tile into LDS ----
    if (tid < 16) lds_r[tid] = r_t[m0 + tid];
    __syncthreads();
    #pragma unroll
    for (int j = 0; j < (16 * kH / 4) / kThreads; ++j) {  // 16 iters, b128 I/O
      const int i4   = tid + j * kThreads;
      const int row  = i4 >> 7;          // / (512/4)
      const int col4 = (i4 & 127) * 4;
      const int brow = m0 + row;
      const v4f x4 = *(const v4f*)&x_t[(size_t)brow * kH + col4];
      const v4f h4 = *(const v4f*)&h_prev[(size_t)brow * kH + col4];
      lds_x4[i4] = x4;
      lds_h4[i4] = lds_r[row] ? v4f{0.0f, 0.0f, 0.0f, 0.0f} : h4;
    }
    __syncthreads();

    // ---- fused dual-GEMM, split accumulators: 8 independent WMMA chains ----
    v8f accx[4] = {v8f{}, v8f{}, v8f{}, v8f{}};
    v8f acch[4] = {v8f{}, v8f{}, v8f{}, v8f{}};

    #pragma unroll 2
    for (int k8 = 0; k8 < kH / 8; ++k8) {  // two WMMA k-steps per iteration
      const int arow = k8 * 8 + 2 * kh;
      // A frags: a[v] = A[lane%16][k + 2*(lane/16) + v]
      const v2f ax0 = *(const v2f*)&lds_x[am * kH + arow];
      const v2f ax1 = *(const v2f*)&lds_x[am * kH + arow + 4];
      const v2f ah0 = *(const v2f*)&lds_h[am * kH + arow];
      const v2f ah1 = *(const v2f*)&lds_h[am * kH + arow + 4];

      #pragma unroll
      for (int g = 0; g < 4; ++g) {
        const v4f wi4 = *(const v4f*)(pwi + k8 * 512 + g * 128);
        const v4f wh4 = *(const v4f*)(pwh + k8 * 512 + g * 128);
        const v2f bi0 = {wi4.x, wi4.y}, bi1 = {wi4.z, wi4.w};
        const v2f bh0 = {wh4.x, wh4.y}, bh1 = {wh4.z, wh4.w};
        accx[g] = __builtin_amdgcn_wmma_f32_16x16x4_f32(
            false, ax0, false, bi0, (short)0, accx[g], false, false);
        acch[g] = __builtin_amdgcn_wmma_f32_16x16x4_f32(
            false, ah0, false, bh0, (short)0, acch[g], false, false);
        accx[g] = __builtin_amdgcn_wmma_f32_16x16x4_f32(
            false, ax1, false, bi1, (short)0, accx[g], false, false);
        acch[g] = __builtin_amdgcn_wmma_f32_16x16x4_f32(
            false, ah1, false, bh1, (short)0, acch[g], false, false);
      }
    }

    // ---- gate nonlinearity + carry update (all four gates in-register) ----
    const int col = j0 + am;
    const float bi_ = bias[0 * kH + col];
    const float bf_ = bias[1 * kH + col];
    const float bg_ = bias[2 * kH + col];
    const float bo_ = bias[3 * kH + col];

    #pragma unroll
    for (int r = 0; r < 8; ++r) {
      const int lrow = r + 8 * kh;       // D row within tile
      const int brow = m0 + lrow;
      const float iv = sigmoid_f(accx[0][r] + acch[0][r] + bi_);
      const float fv = sigmoid_f(accx[1][r] + acch[1][r] + bf_);
      const float gv = fast_tanh(accx[2][r] + acch[2][r] + bg_);
      const float ov = sigmoid_f(accx[3][r] + acch[3][r] + bo_);
      const float cp = lds_r[lrow] ? 0.0f : c_prev[(size_t)brow * kH + col];
      const float cn = fv * cp + iv * gv;
      const float hn = ov * fast_tanh(cn);
      c_ws[(size_t)brow * kH + col]  = cn;
      h_out[(size_t)brow * kH + col] = hn;
    }

    // ---- device-wide barrier: next step reads this step's h (ys[t]) ----
    __threadfence();    // make h/c stores visible at agent scope
    __syncthreads();    // all waves in block done storing / done with lds_r
    if (tid == 0) {
      __hip_atomic_fetch_add(arrive, 1, __ATOMIC_RELEASE,
                             __HIP_MEMORY_SCOPE_AGENT);
    }
    const int target = (t + 1) * kBlocks;  // monotone counter, no reset race
    while (__hip_atomic_load(arrive, __ATOMIC_ACQUIRE,
                             __HIP_MEMORY_SCOPE_AGENT) < target) {
      __builtin_amdgcn_s_sleep(1);
    }
    __syncthreads();
  }
}

}  // namespace

extern "C" void kernel_launch(void* const* d_in, const int* in_sizes, int n_in,
                              void* d_out, int out_size, void* d_ws,
                              size_t ws_size, hipStream_t stream) {
  const float* h0     = (const float*)d_in[0];
  const float* c0     = (const float*)d_in[1];
  const float* ins    = (const float*)d_in[2];
  const int*   resets = (const int*)d_in[3];
  const float* Wi     = (const float*)d_in[4];
  const float* Wh     = (const float*)d_in[5];
  const float* bias   = (const float*)d_in[6];

  float* out = (float*)d_out;
  float* h_f = out;                          // [B,H]
  float* c_f = out + (size_t)kB * kH;        // [B,H]
  float* ys  = out + (size_t)2 * kB * kH;    // [T,B,H]

  // ws layout: c[B*H] f32 | arrive (int, 256B-aligned slot) | pWi | pWh
  char* ws = (char*)d_ws;
  float* c_ws   = (float*)ws;
  int*   arrive = (int*)(ws + (size_t)kB * kH * sizeof(float));
  float* pWi    = (float*)(ws + (size_t)kB * kH * sizeof(float) + 256);
  float* pWh    = pWi + (size_t)kPackElems;

  hipMemsetAsync(arrive, 0, sizeof(int), stream);

  pack_weights_kernel<<<(kPackElems + 255) / 256, 256, 0, stream>>>(
      Wi, Wh, pWi, pWh);

  lstm_scan_kernel<<<kBlocks, kThreads, 0, stream>>>(
      h0, c0, ins, resets, pWi, pWh, bias, ys, c_ws, arrive);

  // h_f = ys[T-1], c_f = final carry
  hipMemcpyAsync(h_f, ys + (size_t)(kT - 1) * kB * kH,
                 (size_t)kB * kH * sizeof(float), hipMemcpyDeviceToDevice,
                 stream);
  hipMemcpyAsync(c_f, c_ws, (size_t)kB * kH * sizeof(float),
                 hipMemcpyDeviceToDevice, stream);
}